// DE_60670708023312
// MI455X (gfx1250) — compile-verified
//
#include <hip/hip_runtime.h>
#include <math.h>

// Problem constants (fixed by the reference).
#define BB      16
#define CM      2048
#define NTOT    4096            // C_M + T_M
#define BN_TOT  (BB * NTOT)     // 65536
#define TPB     256
#define CHUNKS  (NTOT / TPB)    // 16 chunks per batch; 0..7 context, 8..15 target
#define EPS_D   2e-6f
#define BN_EPS  1e-3f
#define BIGF    3.0e38f

typedef float v2f __attribute__((ext_vector_type(2)));
typedef float v8f __attribute__((ext_vector_type(8)));

// ---- CDNA5 async global->LDS staging (ASYNCcnt path, gfx1250) ----
__device__ __forceinline__ void async_copy_b128(unsigned lds_byte_off, const float* gaddr) {
  asm volatile("global_load_async_to_lds_b128 %0, %1, off"
               :: "v"(lds_byte_off), "v"(gaddr)
               : "memory");
}
__device__ __forceinline__ void wait_asynccnt0() {
  asm volatile("s_wait_asynccnt 0" ::: "memory");
}

// Dual-min update; strict '<' keeps lowest index on value ties (matches top_k).
#define UPDT(M1, I1, M2, I2, XC, JJ)                  \
  {                                                   \
    float d_ = fabsf(xq - (XC));                      \
    bool lt1_ = d_ < (M1);                            \
    bool lt2_ = d_ < (M2);                            \
    M2 = lt2_ ? (lt1_ ? (M1) : d_) : (M2);            \
    I2 = lt2_ ? (lt1_ ? (I1) : (JJ)) : (I2);          \
    M1 = lt1_ ? d_ : (M1);                            \
    I1 = lt1_ ? (JJ) : (I1);                          \
  }

// Lexicographic (value, index) compare: exact top_k ordering.
__device__ __forceinline__ bool key_lt(float va, int ia, float vb, int ib) {
  return (va < vb) || ((va == vb) && (ia < ib));
}

// Exact merge of two dual-min trackers (disjoint candidate sets) into the first.
__device__ __forceinline__ void merge2(float& a1v, int& a1i, float& a2v, int& a2i,
                                       float b1v, int b1i, float b2v, int b2i) {
  const bool aFirst = key_lt(a1v, a1i, b1v, b1i);
  const float f1v = aFirst ? a1v : b1v; const int f1i = aFirst ? a1i : b1i;
  const float c1v = aFirst ? a2v : b2v; const int c1i = aFirst ? a2i : b2i; // winner's runner-up
  const float c2v = aFirst ? b1v : a1v; const int c2i = aFirst ? b1i : a1i; // loser's best
  const bool cFirst = key_lt(c1v, c1i, c2v, c2i);
  a1v = f1v; a1i = f1i;
  a2v = cFirst ? c1v : c2v;
  a2i = cFirst ? c1i : c2i;
}

__global__ __launch_bounds__(TPB) void nn_kernel(const float* __restrict__ x_all,
                                                 const float* __restrict__ y_all,
                                                 float* __restrict__ out,
                                                 float* __restrict__ ws) {
  __shared__ __align__(16) float sx[NTOT];
  __shared__ float spart[32];   // 8 waves x 4 WMMA-reduced partials

  const int tid   = threadIdx.x;
  const int blk   = blockIdx.x;
  const int b     = blk >> 4;            // blk / CHUNKS
  const int chunk = blk & 15;            // blk % CHUNKS
  const int n     = chunk * TPB + tid;   // query's pool index within batch
  const bool is_ctx = (chunk < 8);

  const float* xrow = x_all + (size_t)b * NTOT;
  const float* yrow = y_all + (size_t)b * NTOT;

  // Stage candidate pool into LDS: 2 (context) or 4 (target) straight-line async issues.
  const unsigned sbase = (unsigned)(size_t)(&sx[0]);   // low 32 bits == LDS byte offset
  async_copy_b128(sbase + (unsigned)(tid * 16),            xrow + tid * 4);
  async_copy_b128(sbase + (unsigned)((tid + 256) * 16),    xrow + (tid + 256) * 4);
  if (!is_ctx) {
    async_copy_b128(sbase + (unsigned)((tid + 512) * 16),  xrow + (tid + 512) * 4);
    async_copy_b128(sbase + (unsigned)((tid + 768) * 16),  xrow + (tid + 768) * 4);
  }
  wait_asynccnt0();   // this wave's async copies complete
  __syncthreads();    // all waves' copies visible

  // Context queries scan [0,CM); target query n scans [0,n] (penalty-equivalent:
  // >=2049 unpenalized candidates always exist with distance << 1000).
  const int lim = is_ctx ? CM : (n + 1);
  const float xq = sx[n];

  // 4 independent trackers -> breaks the loop-carried select chain (ILP x4).
  float a1 = BIGF, a2 = BIGF, b1 = BIGF, b2 = BIGF;
  float c1 = BIGF, c2 = BIGF, d1m = BIGF, d2m = BIGF;
  int ai1 = 0, ai2 = 0, bi1 = 0, bi2 = 0;
  int ci1 = 0, ci2 = 0, di1 = 0, di2 = 0;

  const int full4 = lim & ~3;
  for (int j = 0; j < full4; j += 4) {
    float4 c = *reinterpret_cast<const float4*>(&sx[j]);  // LDS broadcast read
    UPDT(a1, ai1, a2, ai2, c.x, j + 0);
    UPDT(b1, bi1, b2, bi2, c.y, j + 1);
    UPDT(c1, ci1, c2, ci2, c.z, j + 2);
    UPDT(d1m, di1, d2m, di2, c.w, j + 3);
  }
  for (int j = full4; j < lim; ++j) {    // tail (<=3 iters); indices exceed all others
    UPDT(a1, ai1, a2, ai2, sx[j], j);
  }

  // Exact merges (lexicographic (value,index) => identical result to one serial scan).
  merge2(a1, ai1, a2, ai2, b1, bi1, b2, bi2);
  merge2(c1, ci1, c2, ci2, d1m, di1, d2m, di2);
  merge2(a1, ai1, a2, ai2, c1, ci1, c2, ci2);
  const int ix = ai2;   // second-nearest index

  // Gather neighbor, compute outputs.
  const float cx = sx[ix];
  const float cy = yrow[ix];
  const float yq = yrow[n];
  const float xr = xq - cx;
  const float yr = yq - cy;
  const float d  = yr / (EPS_D + fabsf(xr));
  const float dn = (d != d) ? 10000.0f : d;              // NaN -> 1e4
  const float d2 = (fabsf(d) > 200.0f) ? 0.0f : d;
  const float lab = (d2 == dn) ? 1.0f : 0.0f;

  const int bn = b * NTOT + n;
  out[bn]                      = yr;   // y_diff
  out[BN_TOT + bn]             = xr;   // x_diff
  out[2 * BN_TOT + 2 * bn + 1] = lab;  // d_out[...,1] label
  out[4 * BN_TOT + bn]         = cx;   // x_n
  out[5 * BN_TOT + bn]         = cy;   // y_n

  // ---- BN partial sums via V_WMMA_F32_16X16X4_F32 (exact f32) ----
  // A (16x4, 2 VGPRs): lane l<16 -> (M=l, K0)=d2, (M=l, K1)=d2^2;
  //                    lane l>=16 -> (M=l-16, K2)=d2, (M=l-16, K3)=d2^2.
  // B (4x16, 2 VGPRs): VGPR0 = rows {0,2}, VGPR1 = rows {1,3}.
  //   ones at (K0,N=0),(K2,N=0) select d2; (K1,N=1),(K3,N=1) select d2^2.
  // => D[i][0] = d2_i + d2_{i+16};  D[i][1] = q_i + q_{i+16}.
  const int lane = tid & 31;
  const int wv   = tid >> 5;
  v2f amat; amat.x = d2; amat.y = d2 * d2;
  v2f bsel;
  bsel.x = ((lane & 15) == 0) ? 1.0f : 0.0f;
  bsel.y = ((lane & 15) == 1) ? 1.0f : 0.0f;
  v8f acc = {};
  acc = __builtin_amdgcn_wmma_f32_16x16x4_f32(false, amat, false, bsel,
                                              (short)0, acc, false, false);
  // Lane = column: lanes {0,16} hold the 16 d2-pair-sums (8 rows each),
  // lanes {1,17} hold the d2^2 pair-sums. Sum each stripe of 8.
  const float part = acc[0] + acc[1] + acc[2] + acc[3] +
                     acc[4] + acc[5] + acc[6] + acc[7];
  if ((lane & 15) < 2) {
    spart[wv * 4 + (lane & 15) * 2 + (lane >> 4)] = part;
  }
  __syncthreads();
  if (tid < 2) {   // tid==0 -> sum(d2), tid==1 -> sum(d2^2)
    float s = 0.0f;
    for (int w = 0; w < 8; ++w) {
      s += spart[w * 4 + tid * 2 + 0] + spart[w * 4 + tid * 2 + 1];
    }
    ws[tid * 256 + blk] = s;   // ws[blk]=sum d2, ws[256+blk]=sum d2^2
  }
}

__global__ __launch_bounds__(256) void reduce_kernel(float* __restrict__ ws) {
  __shared__ float s1[256];
  __shared__ float s2[256];
  const int t = threadIdx.x;
  s1[t] = ws[t];
  s2[t] = ws[256 + t];
  __syncthreads();
  for (int s = 128; s > 0; s >>= 1) {
    if (t < s) { s1[t] += s1[t + s]; s2[t] += s2[t + s]; }
    __syncthreads();
  }
  if (t == 0) {
    const float cnt  = (float)BN_TOT;
    const float mean = s1[0] / cnt;
    const float var  = s2[0] / cnt - mean * mean;
    ws[512] = mean;
    ws[513] = rsqrtf(var + BN_EPS);
  }
}

__global__ __launch_bounds__(256) void bn_kernel(float* __restrict__ out,
                                                 const float* __restrict__ ws,
                                                 const float* __restrict__ gamma,
                                                 const float* __restrict__ beta) {
  const int i = blockIdx.x * 256 + threadIdx.x;   // 0 .. BN_TOT-1
  const float yr = out[i];                 // y_diff (already written by nn_kernel)
  const float xr = out[BN_TOT + i];        // x_diff
  const float d  = yr / (EPS_D + fabsf(xr));
  const float d2 = (fabsf(d) > 200.0f) ? 0.0f : d;
  const float mean   = ws[512];
  const float invstd = ws[513];
  out[2 * BN_TOT + 2 * i] = gamma[0] * (d2 - mean) * invstd + beta[0];
}

extern "C" void kernel_launch(void* const* d_in, const int* in_sizes, int n_in,
                              void* d_out, int out_size, void* d_ws, size_t ws_size,
                              hipStream_t stream) {
  (void)in_sizes; (void)n_in; (void)out_size; (void)ws_size;
  const float* y_all = (const float*)d_in[0];
  const float* x_all = (const float*)d_in[1];
  const float* gamma = (const float*)d_in[8];
  const float* beta  = (const float*)d_in[9];
  float* out = (float*)d_out;
  float* ws  = (float*)d_ws;

  nn_kernel<<<BB * CHUNKS, TPB, 0, stream>>>(x_all, y_all, out, ws);
  reduce_kernel<<<1, 256, 0, stream>>>(ws);
  bn_kernel<<<BN_TOT / 256, 256, 0, stream>>>(out, ws, gamma, beta);
}